// ReLU_Autoencoder_31318901522868
// MI455X (gfx1250) — compile-verified
//
#include <hip/hip_runtime.h>
#include <math.h>

// Problem dims (fixed by reference)
#define B_DIM 512
#define P_DIM 1024
#define N_DIM 1024
#define T_DIM 32
#define EPS_F 1e-10f
#define MLE_STEPS 15
#define MLE_LR 0.01f

// GEMM blocking: block = 256 threads = 8 waves.
// Macro-tile: 128 rows (one 16-row tile per wave) x 64 cols (4 WMMA tiles/wave).
// K staged through LDS in chunks of 16, double buffered.
#define KCHUNK 16
#define NSTRIP 64

typedef float v2f __attribute__((ext_vector_type(2)));
typedef float v4f __attribute__((ext_vector_type(4)));
typedef float v8f __attribute__((ext_vector_type(8)));
typedef int v4i __attribute__((vector_size(16)));  // matches builtin signature

#define GLOBAL_AS __attribute__((address_space(1)))
#define LDS_AS __attribute__((address_space(3)))

#if defined(__has_builtin)
#if __has_builtin(__builtin_amdgcn_global_load_async_to_lds_b128) && \
    __has_builtin(__builtin_amdgcn_s_wait_asynccnt)
#define USE_ASYNC_LDS 1
#endif
#endif

__device__ __forceinline__ float wave_sum(float v) {
#pragma unroll
  for (int off = 16; off >= 1; off >>= 1) v += __shfl_xor(v, off, 32);
  return v;
}
__device__ __forceinline__ float wave_max(float v) {
#pragma unroll
  for (int off = 16; off >= 1; off >>= 1) v = fmaxf(v, __shfl_xor(v, off, 32));
  return v;
}

// Stage one KCHUNK x NSTRIP chunk of the B matrix into LDS.
// 256 threads: thread i moves 16 bytes: row = i/16, cols (i%16)*4 .. +3.
__device__ __forceinline__ void stage_b_chunk(const float* __restrict__ w,
                                              int k0, int n0, int ldw,
                                              float* lds_dst, int tid) {
  const int row = tid >> 4;        // 0..15
  const int c4 = (tid & 15) << 2;  // 0,4,...,60
  float* src = const_cast<float*>(w + (size_t)(k0 + row) * ldw + n0 + c4);
  float* dst = lds_dst + row * NSTRIP + c4;
#if defined(USE_ASYNC_LDS)
  __builtin_amdgcn_global_load_async_to_lds_b128(
      (GLOBAL_AS v4i*)src, (LDS_AS v4i*)dst, 0, 0);
#else
  *(v4f*)dst = *(const v4f*)src;
#endif
}

__device__ __forceinline__ void wait_async_1() {
#if defined(USE_ASYNC_LDS)
  __builtin_amdgcn_s_wait_asynccnt(1);
#endif
}
__device__ __forceinline__ void wait_async_0() {
#if defined(USE_ASYNC_LDS)
  __builtin_amdgcn_s_wait_asynccnt(0);
#endif
}

// 4 K-steps (k=4 each) of WMMA over a staged B chunk; A fragment reused for 4 N-tiles.
__device__ __forceinline__ void mma_chunk(const float* __restrict__ arow,
                                          int kbase, const float* ldsB,
                                          int l16, int lh, v8f acc[4]) {
#pragma unroll
  for (int kk = 0; kk < KCHUNK; kk += 4) {
    v2f av = *(const v2f*)(arow + kbase + kk);  // A[m][k+2lh], A[m][k+2lh+1]
    const float* brow = ldsB + (kk + 2 * lh) * NSTRIP;
#pragma unroll
    for (int c = 0; c < 4; ++c) {
      v2f bv;
      bv.x = brow[c * 16 + l16];           // B[k+2lh  ][n]
      bv.y = brow[NSTRIP + c * 16 + l16];  // B[k+2lh+1][n]
      acc[c] = __builtin_amdgcn_wmma_f32_16x16x4_f32(
          false, av, false, bv, (short)0, acc[c], false, false);
    }
  }
}

// ---------------------------------------------------------------------------
// K1: a = relu(x @ w_enc + b_enc); a_sig = 2/(1+exp(-a)) - 1
// ---------------------------------------------------------------------------
__global__ __launch_bounds__(256) void enc_gemm_kernel(
    const float* __restrict__ x, const float* __restrict__ w_enc,
    const float* __restrict__ b_enc, float* __restrict__ a_out,
    float* __restrict__ asig_out) {
  __shared__ float ldsB[2][KCHUNK * NSTRIP];
  const int tid = threadIdx.x;
  const int waveId = tid >> 5;
  const int lane = tid & 31;
  const int l16 = lane & 15;
  const int lh = lane >> 4;

  const int bn = blockIdx.x & 15;  // N/64 = 16 strips
  const int bm = blockIdx.x >> 4;  // B/128 = 4
  const int n0 = bn * NSTRIP;
  const int m = bm * 128 + waveId * 16 + l16;

  const float* arow = x + (size_t)m * P_DIM + 2 * lh;
  v8f acc[4];
#pragma unroll
  for (int c = 0; c < 4; ++c) acc[c] = (v8f){0.f,0.f,0.f,0.f,0.f,0.f,0.f,0.f};

  const int NCH = P_DIM / KCHUNK;  // 64 chunks
  stage_b_chunk(w_enc, 0, n0, N_DIM, &ldsB[0][0], tid);
  for (int c = 0; c < NCH - 1; ++c) {
    stage_b_chunk(w_enc, (c + 1) * KCHUNK, n0, N_DIM, &ldsB[(c + 1) & 1][0], tid);
    __builtin_prefetch(arow + (c + 1) * KCHUNK, 0, 3);
    wait_async_1();
    __syncthreads();
    mma_chunk(arow, c * KCHUNK, &ldsB[c & 1][0], l16, lh, acc);
    __syncthreads();
  }
  wait_async_0();
  __syncthreads();
  mma_chunk(arow, (NCH - 1) * KCHUNK, &ldsB[(NCH - 1) & 1][0], l16, lh, acc);

  const int row_base = bm * 128 + waveId * 16 + lh * 8;  // lanes 16-31 -> M+8
#pragma unroll
  for (int c = 0; c < 4; ++c) {
    const int nn = n0 + c * 16 + l16;
    const float bias = b_enc[nn];
#pragma unroll
    for (int r = 0; r < 8; ++r) {
      const int row = row_base + r;
      float av = fmaxf(acc[c][r] + bias, 0.0f);
      const float sg = 2.0f / (1.0f + expf(-av)) - 1.0f;
      a_out[(size_t)row * N_DIM + nn] = av;
      asig_out[(size_t)row * N_DIM + nn] = sg;
    }
  }
}

// ---------------------------------------------------------------------------
// K2: per-neuron MLE gradient ascent + entropy. One wave per neuron.
// Triangle basis is 2-sparse: p = (1-f)*pi[j] + f*pi[j+1].
// grad_t = pi_t * ( sum_b phi_t/(p+eps) - sum_b p/(p+eps) )
// ---------------------------------------------------------------------------
__global__ __launch_bounds__(32) void mle_entropy_kernel(
    const float* __restrict__ asig, const float* __restrict__ thetas,
    const float* __restrict__ centers, float* __restrict__ ent_out) {
  __shared__ float pi_s[T_DIM];
  __shared__ float s_s[T_DIM];
  __shared__ float f_s[B_DIM];
  __shared__ int j_s[B_DIM];

  const int n = blockIdx.x;
  const int t = threadIdx.x;

  float th = thetas[n * T_DIM + t];
  const float c0 = centers[0];
  const float invw = 1.0f / (centers[1] - centers[0]);

  for (int b = t; b < B_DIM; b += 32) {
    const float as = asig[(size_t)b * N_DIM + n];
    const float u = (as - c0) * invw;
    int j = (int)floorf(u);
    j = max(0, min(j, T_DIM - 2));
    f_s[b] = u - (float)j;
    j_s[b] = j;
  }
  __syncthreads();

  for (int step = 0; step < MLE_STEPS; ++step) {
    const float mx = wave_max(th);
    const float e = expf(th - mx);
    const float sm = wave_sum(e);
    const float pi = e / sm;
    pi_s[t] = pi;
    s_s[t] = 0.0f;
    __syncthreads();

    float s0 = 0.0f;
    for (int b = t; b < B_DIM; b += 32) {
      const int j = j_s[b];
      const float f = f_s[b];
      const float p = (1.0f - f) * pi_s[j] + f * pi_s[j + 1];
      const float r = 1.0f / (p + EPS_F);
      atomicAdd(&s_s[j], (1.0f - f) * r);  // ds_add_f32 scatter
      atomicAdd(&s_s[j + 1], f * r);
      s0 += p * r;
    }
    __syncthreads();
    s0 = wave_sum(s0);
    th += MLE_LR * (pi * (s_s[t] - s0));
    __syncthreads();
  }

  const float mx = wave_max(th);
  const float e = expf(th - mx);
  const float sm = wave_sum(e);
  pi_s[t] = e / sm;
  __syncthreads();

  float ent = 0.0f;
  for (int b = t; b < B_DIM; b += 32) {
    const int j = j_s[b];
    const float f = f_s[b];
    const float p = (1.0f - f) * pi_s[j] + f * pi_s[j + 1];
    ent -= p * logf(p + EPS_F);
  }
  ent = wave_sum(ent);
  if (t == 0) ent_out[n] = ent;
}

// ---------------------------------------------------------------------------
// K3: x_hat = a @ w_dec + b_dec; per-wave recon partial sum(d^2)
// ---------------------------------------------------------------------------
__global__ __launch_bounds__(256) void dec_gemm_kernel(
    const float* __restrict__ a, const float* __restrict__ w_dec,
    const float* __restrict__ b_dec, const float* __restrict__ x,
    float* __restrict__ xhat, float* __restrict__ recon_part) {
  __shared__ float ldsB[2][KCHUNK * NSTRIP];
  const int tid = threadIdx.x;
  const int waveId = tid >> 5;
  const int lane = tid & 31;
  const int l16 = lane & 15;
  const int lh = lane >> 4;

  const int bp = blockIdx.x & 15;  // P/64 = 16 strips
  const int bm = blockIdx.x >> 4;  // B/128 = 4
  const int p0 = bp * NSTRIP;
  const int m = bm * 128 + waveId * 16 + l16;

  const float* arow = a + (size_t)m * N_DIM + 2 * lh;
  v8f acc[4];
#pragma unroll
  for (int c = 0; c < 4; ++c) acc[c] = (v8f){0.f,0.f,0.f,0.f,0.f,0.f,0.f,0.f};

  const int NCH = N_DIM / KCHUNK;  // 64 chunks
  stage_b_chunk(w_dec, 0, p0, P_DIM, &ldsB[0][0], tid);
  for (int c = 0; c < NCH - 1; ++c) {
    stage_b_chunk(w_dec, (c + 1) * KCHUNK, p0, P_DIM, &ldsB[(c + 1) & 1][0], tid);
    __builtin_prefetch(arow + (c + 1) * KCHUNK, 0, 3);
    wait_async_1();
    __syncthreads();
    mma_chunk(arow, c * KCHUNK, &ldsB[c & 1][0], l16, lh, acc);
    __syncthreads();
  }
  wait_async_0();
  __syncthreads();
  mma_chunk(arow, (NCH - 1) * KCHUNK, &ldsB[(NCH - 1) & 1][0], l16, lh, acc);

  const int row_base = bm * 128 + waveId * 16 + lh * 8;
  float part = 0.0f;
#pragma unroll
  for (int c = 0; c < 4; ++c) {
    const int pp = p0 + c * 16 + l16;
    const float bias = b_dec[pp];
#pragma unroll
    for (int r = 0; r < 8; ++r) {
      const int row = row_base + r;
      const float xh = acc[c][r] + bias;
      xhat[(size_t)row * P_DIM + pp] = xh;
      const float d = x[(size_t)row * P_DIM + pp] - xh;
      part += d * d;
    }
  }
  part = wave_sum(part);
  if (lane == 0) recon_part[blockIdx.x * 8 + waveId] = part;  // 512 slots
}

// ---------------------------------------------------------------------------
// K4: sum of squares of w_enc and w_dec -> per-block partials
// ---------------------------------------------------------------------------
__global__ __launch_bounds__(256) void sumsq_kernel(
    const float* __restrict__ w_enc, const float* __restrict__ w_dec,
    float* __restrict__ decay_part) {
  __shared__ float red[256];
  const int tid = threadIdx.x;
  const int total = P_DIM * N_DIM;
  float s = 0.0f;
  for (int i = blockIdx.x * 256 + tid; i < total; i += gridDim.x * 256) {
    const float we = w_enc[i];
    const float wd = w_dec[i];
    s += we * we + wd * wd;
  }
  red[tid] = s;
  for (int off = 128; off > 0; off >>= 1) {
    __syncthreads();
    if (tid < off) red[tid] += red[tid + off];
  }
  if (tid == 0) decay_part[blockIdx.x] = red[0];
}

// ---------------------------------------------------------------------------
// K5: deterministic final combine -> total_loss
// ---------------------------------------------------------------------------
__global__ __launch_bounds__(256) void combine_kernel(
    const float* __restrict__ recon_part, const float* __restrict__ decay_part,
    const float* __restrict__ ent, float* __restrict__ total_out) {
  __shared__ float red[256];
  const int tid = threadIdx.x;

  float r = 0.0f;
  for (int i = tid; i < 512; i += 256) r += recon_part[i];
  red[tid] = r;
  for (int off = 128; off > 0; off >>= 1) {
    __syncthreads();
    if (tid < off) red[tid] += red[tid + off];
  }
  __syncthreads();
  const float rsum = red[0];
  __syncthreads();

  red[tid] = decay_part[tid];
  for (int off = 128; off > 0; off >>= 1) {
    __syncthreads();
    if (tid < off) red[tid] += red[tid + off];
  }
  __syncthreads();
  const float dsum = red[0];
  __syncthreads();

  float en = 0.0f;
  for (int i = tid; i < N_DIM; i += 256) en += ent[i];
  red[tid] = en;
  for (int off = 128; off > 0; off >>= 1) {
    __syncthreads();
    if (tid < off) red[tid] += red[tid + off];
  }
  __syncthreads();
  const float esum = red[0];

  if (tid == 0) {
    const float recon_loss = (0.5f / (float)B_DIM) * rsum;
    const float ent_loss = 0.01f * esum;
    const float decay_loss = 0.5f * 0.001f * 0.5f * dsum;  // 0.00025*(Se+Sd)
    total_out[0] = recon_loss + ent_loss + decay_loss;
  }
}

// ---------------------------------------------------------------------------
extern "C" void kernel_launch(void* const* d_in, const int* in_sizes, int n_in,
                              void* d_out, int out_size, void* d_ws,
                              size_t ws_size, hipStream_t stream) {
  const float* x       = (const float*)d_in[0];  // [512,1024]
  const float* w_enc   = (const float*)d_in[1];  // [1024,1024]
  const float* b_enc   = (const float*)d_in[2];  // [1024]
  const float* w_dec   = (const float*)d_in[3];  // [1024,1024]
  const float* b_dec   = (const float*)d_in[4];  // [1024]
  const float* thetas  = (const float*)d_in[5];  // [1024,32]
  const float* centers = (const float*)d_in[6];  // [32]

  float* out   = (float*)d_out;
  float* xhat  = out;                          // 512*1024
  float* ent   = out + (size_t)B_DIM * P_DIM;  // 1024
  float* total = ent + N_DIM;                  // 1

  float* ws         = (float*)d_ws;
  float* a_buf      = ws;                                // 512*1024
  float* asig_buf   = a_buf + (size_t)B_DIM * N_DIM;     // 512*1024
  float* recon_part = asig_buf + (size_t)B_DIM * N_DIM;  // 512
  float* decay_part = recon_part + 512;                  // 256

  // Encoder GEMM + relu + sigmoid. 64 blocks x 8 waves, LDS-staged B.
  enc_gemm_kernel<<<64, 256, 0, stream>>>(x, w_enc, b_enc, a_buf, asig_buf);

  // Per-neuron MLE (15 steps) + entropy. One wave per neuron.
  mle_entropy_kernel<<<N_DIM, 32, 0, stream>>>(asig_buf, thetas, centers, ent);

  // Decoder GEMM + bias + recon partials.
  dec_gemm_kernel<<<64, 256, 0, stream>>>(a_buf, w_dec, b_dec, x, xhat,
                                          recon_part);

  // Weight-decay sum of squares.
  sumsq_kernel<<<256, 256, 0, stream>>>(w_enc, w_dec, decay_part);

  // Final deterministic combine.
  combine_kernel<<<1, 256, 0, stream>>>(recon_part, decay_part, ent, total);
}